// FusedScaleMaskSoftmax_33792802685307
// MI455X (gfx1250) — compile-verified
//
#include <hip/hip_runtime.h>

// FusedScaleMaskSoftmax: x[2,16,2048,2048] fp16, mask[2,1,2048,2048] bool,
// out fp16, softmax in fp32, scale=0.125, masked_fill=-10000.
//
// One wave32 per row (64 elems/lane), 8 rows per 256-thread block.
// Pass 1 runs in packed fp16 (v_perm / v_pk_mul_lo_u16 / v_pk_add_f16 /
// v_pk_max_num_f16): mask bytes (0/1) become an additive {0, -inf} penalty,
// so masked lanes vanish from the row max and exp exactly like exp(-10000).
// Mask row staged via CDNA5 async global->LDS; x streamed via b128.

#define SK      2048
#define SQ      2048
#define NHEADS  16
#define WARPS   8            // rows per block
#define THREADS (WARPS * 32)
#define NCHUNK  8            // 8 chunks x 4 dwords (8 halves) per lane

#define LOG2E     1.44269504088896340736f
#define SCALE_L2E (0.125f * LOG2E)

typedef __attribute__((ext_vector_type(2))) _Float16       h2;
typedef __attribute__((ext_vector_type(2))) float          f32x2;
typedef __attribute__((ext_vector_type(2))) unsigned short u16x2;
typedef __attribute__((ext_vector_type(2))) unsigned int   u32x2;
typedef __attribute__((ext_vector_type(4))) unsigned int   u32x4;

#if __has_builtin(__builtin_amdgcn_exp2f)
#define EXP2F(x) __builtin_amdgcn_exp2f(x)
#else
#define EXP2F(x) exp2f(x)
#endif

__global__ __launch_bounds__(THREADS) void fused_scale_mask_softmax_kernel(
    const _Float16* __restrict__ x,
    const unsigned char* __restrict__ mask,   // jnp.bool_ -> 1 byte, 0/1
    _Float16* __restrict__ out) {
  __shared__ unsigned char smask[WARPS][SK];  // per-wave mask row (2 KB)

  const int tid  = threadIdx.x;
  const int lane = tid & 31;
  const int w    = tid >> 5;

  const long row = (long)blockIdx.x * WARPS + w;        // [0, B*H*SQ)
  const long b   = row / ((long)NHEADS * SQ);
  const long sq  = row & (SQ - 1);

  const _Float16* xrow = x + row * (long)SK;
  _Float16* orow       = out + row * (long)SK;
  const unsigned char* mrow = mask + (b * SQ + sq) * (long)SK;

  // ---- CDNA5 async global->LDS: stage this wave's 2KB mask row ----
#pragma unroll
  for (int k = 0; k < 4; ++k) {
    unsigned int lds_off =
        (unsigned int)(unsigned long long)(&smask[w][512 * k + 16 * lane]);
    unsigned long long ga = (unsigned long long)(mrow + 512 * k + 16 * lane);
    asm volatile("global_load_async_to_lds_b128 %0, %1, off"
                 :
                 : "v"(lds_off), "v"(ga)
                 : "memory");
  }

  // ---- x: 8 coalesced b128 loads (contiguous 512B per instruction) ----
  u32x4 xw[NCHUNK];
#pragma unroll
  for (int j = 0; j < NCHUNK; ++j)
    xw[j] = *(const u32x4*)(xrow + 512 * j + 8 * lane);

  // Only this wave touches smask[w]: asynccnt==0 is a sufficient fence.
  asm volatile("s_wait_asynccnt 0" ::: "memory");

  // ---- pass 1: packed fp16 mask-apply + packed fp16 row max ----
  h2 xm2[NCHUNK * 4];
  h2 mx2 = __builtin_bit_cast(h2, 0xFC00FC00u);   // {-inf, -inf}
#pragma unroll
  for (int j = 0; j < NCHUNK; ++j) {
    const u32x2 mw = *(const u32x2*)(&smask[w][512 * j + 8 * lane]);
#pragma unroll
    for (int p = 0; p < 4; ++p) {
      // Spread the 2 mask bytes of this half-pair into u16 lanes (v_perm_b32).
      const unsigned int sel = (p & 1) ? 0x04030402u : 0x04010400u;
      unsigned int t = __builtin_amdgcn_perm(0u, mw[p >> 1], sel);
      // 0/1 per u16 lane -> {0x0000, 0xFC00(-inf)} (v_pk_mul_lo_u16)
      u16x2 pen = __builtin_bit_cast(u16x2, t) * (unsigned short)0xFC00u;
      // x + {0,-inf}: exact passthrough or -inf (v_pk_add_f16)
      h2 xm = __builtin_bit_cast(h2, xw[j][p]) + __builtin_bit_cast(h2, pen);
      xm2[4 * j + p] = xm;
      mx2 = __builtin_elementwise_max(mx2, xm);   // v_pk_max_num_f16
    }
  }
  // fold packed max -> fp32, scale (max commutes with positive scale)
  float M = fmaxf((float)mx2.x, (float)mx2.y) * SCALE_L2E;
#pragma unroll
  for (int off = 16; off > 0; off >>= 1)
    M = fmaxf(M, __shfl_xor(M, off, 32));

  // ---- pass 2: e = 2^(x*S - M) via v_exp_f32; packed f32 sum ----
  f32x2 e2[NCHUNK * 4];
  f32x2 a0 = {0.f, 0.f}, a1 = {0.f, 0.f};
#pragma unroll
  for (int k = 0; k < NCHUNK * 4; ++k) {
    float glo = (float)xm2[k].x * SCALE_L2E - M;  // fma (mix) after cvt
    float ghi = (float)xm2[k].y * SCALE_L2E - M;
    f32x2 e;
    e.x = EXP2F(glo);
    e.y = EXP2F(ghi);
    e2[k] = e;
    if (k & 1) a1 += e; else a0 += e;             // v_pk_add_f32
  }
  f32x2 a = a0 + a1;
  float s = a.x + a.y;
#pragma unroll
  for (int off = 16; off > 0; off >>= 1)
    s += __shfl_xor(s, off, 32);

  // ---- pass 3: normalize (v_pk_mul_f32), pack fp16, b128 stores ----
  const float inv = __builtin_amdgcn_rcpf(s);
#pragma unroll
  for (int j = 0; j < NCHUNK; ++j) {
    u32x4 ow;
#pragma unroll
    for (int p = 0; p < 4; ++p) {
      f32x2 r = e2[4 * j + p] * inv;
      h2 o;
      o.x = (_Float16)r.x;
      o.y = (_Float16)r.y;
      ow[p] = __builtin_bit_cast(unsigned int, o);
    }
    *(u32x4*)(orow + 512 * j + 8 * lane) = ow;
  }
}

extern "C" void kernel_launch(void* const* d_in, const int* in_sizes, int n_in,
                              void* d_out, int out_size, void* d_ws, size_t ws_size,
                              hipStream_t stream) {
  const _Float16* x = (const _Float16*)d_in[0];
  const unsigned char* mask = (const unsigned char*)d_in[1];
  _Float16* out = (_Float16*)d_out;

  const int rows   = in_sizes[0] / SK;     // B*H*SQ = 65536
  const int blocks = rows / WARPS;         // 8192
  fused_scale_mask_softmax_kernel<<<blocks, THREADS, 0, stream>>>(x, mask, out);
}